// QuantumAttention_48507360641846
// MI455X (gfx1250) — compile-verified
//
#include <hip/hip_runtime.h>
#include <math.h>

typedef float v2f __attribute__((ext_vector_type(2)));
typedef float v8f __attribute__((ext_vector_type(8)));

#define IN_DIM 96

// prep buffer layout (floats):
//   [0..31]    UQ: 4 wires x (u00,u01,u10,u11) complex = 8 floats each
//   [32..63]   UK
//   [64..95]   UV
//   [96..127]  UC
//   [128..135] CRXQ trig: 4 x (cos,sin)
//   [136..143] CRXK trig
//   [144..151] CRXV trig
//   [152..167] CRXC trig: 8 x (cos,sin)
#define P_UQ   0
#define P_UK   32
#define P_UV   64
#define P_UC   96
#define P_CRXQ 128
#define P_CRXK 136
#define P_CRXV 144
#define P_CRXC 152
#define P_N    168

// =====================================================================
// Kernel 0: precompute batch-invariant gate algebra.
// U = Rz(g)*Rx(b)*Rz(a):
//   u00 = cb*e^{-i(a+g)/2}, u01 = sb*(sinM - i cosM), M=(a-g)/2
//   u10 = sb*(-sinM - i cosM), u11 = conj(u00)
// =====================================================================
__global__ void qattn_prep(const float* __restrict__ wq_rot, const float* __restrict__ wq_crx,
                           const float* __restrict__ wk_rot, const float* __restrict__ wk_crx,
                           const float* __restrict__ wv_rot, const float* __restrict__ wv_crx,
                           const float* __restrict__ wc_rot, const float* __restrict__ wc_crx,
                           float* __restrict__ prep) {
  int t = threadIdx.x;
  if (t < 16) {
    int g = t >> 2, w = t & 3;
    const float* rot = (g == 0) ? wq_rot : (g == 1) ? wk_rot : (g == 2) ? wv_rot : wc_rot;
    float a = rot[3*w], bb = rot[3*w + 1], gg = rot[3*w + 2];
    float cb = __cosf(0.5f * bb), sb = __sinf(0.5f * bb);
    float P = 0.5f * (a + gg), M = 0.5f * (a - gg);
    float cP = __cosf(P), sP = __sinf(P);
    float cM = __cosf(M), sM = __sinf(M);
    float* U = prep + 32 * g + 8 * w;
    U[0] =  cb * cP;  U[1] = -cb * sP;   // u00
    U[2] =  sb * sM;  U[3] = -sb * cM;   // u01
    U[4] = -sb * sM;  U[5] = -sb * cM;   // u10
    U[6] =  cb * cP;  U[7] =  cb * sP;   // u11 = conj(u00)
  } else if (t < 36) {
    int idx = t - 16;   // 0..19
    float ang; int off;
    if      (idx < 4)  { ang = wq_crx[idx];      off = P_CRXQ + 2 * idx; }
    else if (idx < 8)  { ang = wk_crx[idx - 4];  off = P_CRXK + 2 * (idx - 4); }
    else if (idx < 12) { ang = wv_crx[idx - 8];  off = P_CRXV + 2 * (idx - 8); }
    else               { ang = wc_crx[idx - 12]; off = P_CRXC + 2 * (idx - 12); }
    prep[off]     = __cosf(0.5f * ang);
    prep[off + 1] = __sinf(0.5f * ang);
  }
}

// =====================================================================
// Kernel 1: xq = x1 @ W^T + b ; xk = x2 @ W^T + b  via V_WMMA_F32_16X16X4_F32
// =====================================================================
__global__ void qattn_proj(const float* __restrict__ x1, const float* __restrict__ x2,
                           const float* __restrict__ W, const float* __restrict__ bias,
                           float* __restrict__ xq, float* __restrict__ xk) {
  int wave  = threadIdx.x >> 5;
  int lane  = threadIdx.x & 31;
  int tile  = blockIdx.x * (blockDim.x >> 5) + wave;   // 16 samples per tile
  int khalf = lane >> 4;
  long row  = (long)(tile * 16 + (lane & 15));
  int n     = lane & 15;
  int nb    = n & 3;
  bool nvalid = (n < 4);

  v8f cq = {0.f,0.f,0.f,0.f,0.f,0.f,0.f,0.f};
  v8f ck = {0.f,0.f,0.f,0.f,0.f,0.f,0.f,0.f};

  #pragma unroll 4
  for (int k0 = 0; k0 < IN_DIM; k0 += 4) {
    int kb = k0 + 2 * khalf;
    v2f a1, a2, bm;
    a1.x = x1[row * IN_DIM + kb];
    a1.y = x1[row * IN_DIM + kb + 1];
    a2.x = x2[row * IN_DIM + kb];
    a2.y = x2[row * IN_DIM + kb + 1];
    float w0 = W[nb * IN_DIM + kb];
    float w1 = W[nb * IN_DIM + kb + 1];
    bm.x = nvalid ? w0 : 0.f;
    bm.y = nvalid ? w1 : 0.f;
    cq = __builtin_amdgcn_wmma_f32_16x16x4_f32(false, a1, false, bm, (short)0, cq, false, false);
    ck = __builtin_amdgcn_wmma_f32_16x16x4_f32(false, a2, false, bm, (short)0, ck, false, false);
  }

  if (nvalid) {
    float bv = bias[n];
    #pragma unroll
    for (int r = 0; r < 8; ++r) {
      int m = r + 8 * khalf;
      xq[(long)(tile * 16 + m) * 4 + n] = cq[r] + bv;
      xk[(long)(tile * 16 + m) * 4 + n] = ck[r] + bv;
    }
  }
}

// ============ in-lane 16-amp helpers (wire w <-> bit 3-w) ============
template <int C, int T>
__device__ __forceinline__ void crx4(float* vr, float* vi, float c, float s) {
  constexpr int m = 8 >> T;
  #pragma unroll
  for (int r = 0; r < 16; ++r) {
    if (((r & m) == 0) && (((r >> (3 - C)) & 1) != 0)) {
      int q = r | m;
      float a0r = vr[r], a0i = vi[r], a1r = vr[q], a1i = vi[q];
      vr[r] = c * a0r + s * a1i;  vi[r] = c * a0i - s * a1r;
      vr[q] = c * a1r + s * a0i;  vi[q] = c * a1i - s * a0r;
    }
  }
}

template <int C, int T>
__device__ __forceinline__ void cnot4(float* vr, float* vi) {
  constexpr int m = 8 >> T;
  #pragma unroll
  for (int r = 0; r < 16; ++r) {
    if (((r & m) == 0) && (((r >> (3 - C)) & 1) != 0)) {
      int q = r | m;
      float tr = vr[r], ti = vi[r];
      vr[r] = vr[q];  vi[r] = vi[q];
      vr[q] = tr;     vi[q] = ti;
    }
  }
}

// ============ distributed 8-qubit helpers ============================
// a[7:0], wire w <-> bit 7-w; lane = a>>3, reg = a&7.
// wire 0..4 -> lane bit (4-w) / shfl mask 16>>w;  wire 5..7 -> reg bit 7-w
__device__ __forceinline__ float wave_sum(float v) {
  v += __shfl_xor(v, 16, 32);
  v += __shfl_xor(v, 8, 32);
  v += __shfl_xor(v, 4, 32);
  v += __shfl_xor(v, 2, 32);
  v += __shfl_xor(v, 1, 32);
  return v;
}
__device__ __forceinline__ float half_sum(float v) {   // 16-lane halves
  v += __shfl_xor(v, 8, 32);
  v += __shfl_xor(v, 4, 32);
  v += __shfl_xor(v, 2, 32);
  v += __shfl_xor(v, 1, 32);
  return v;
}

template <int W>
__device__ __forceinline__ int bit8(int lane, int r) {
  if constexpr (W <= 4) return (lane >> (4 - W)) & 1;
  else                  return (r    >> (7 - W)) & 1;
}

template <int CW, int TW>
__device__ __forceinline__ void crx8(float* sr, float* si, float c, float s, int lane) {
  if constexpr (TW <= 4) {
    constexpr int mask = 16 >> TW;
    #pragma unroll
    for (int r = 0; r < 8; ++r) {
      float pr = __shfl_xor(sr[r], mask, 32);
      float pi = __shfl_xor(si[r], mask, 32);
      bool ctl = bit8<CW>(lane, r) != 0;
      float ar = sr[r], ai = si[r];
      sr[r] = ctl ? (c * ar + s * pi) : ar;
      si[r] = ctl ? (c * ai - s * pr) : ai;
    }
  } else {
    constexpr int m = 1 << (7 - TW);
    #pragma unroll
    for (int r = 0; r < 8; ++r) {
      if ((r & m) == 0) {
        int q = r | m;
        bool ctl = bit8<CW>(lane, r) != 0;
        float a0r = sr[r], a0i = si[r], a1r = sr[q], a1i = si[q];
        float n0r = c * a0r + s * a1i, n0i = c * a0i - s * a1r;
        float n1r = c * a1r + s * a0i, n1i = c * a1i - s * a0r;
        sr[r] = ctl ? n0r : a0r;  si[r] = ctl ? n0i : a0i;
        sr[q] = ctl ? n1r : a1r;  si[q] = ctl ? n1i : a1i;
      }
    }
  }
}

template <int CW, int TW>
__device__ __forceinline__ void cnot8(float* sr, float* si, int lane) {
  if constexpr (TW <= 4) {
    constexpr int mask = 16 >> TW;
    #pragma unroll
    for (int r = 0; r < 8; ++r) {
      float pr = __shfl_xor(sr[r], mask, 32);
      float pi = __shfl_xor(si[r], mask, 32);
      bool ctl = bit8<CW>(lane, r) != 0;
      sr[r] = ctl ? pr : sr[r];
      si[r] = ctl ? pi : si[r];
    }
  } else {
    constexpr int m = 1 << (7 - TW);
    #pragma unroll
    for (int r = 0; r < 8; ++r) {
      if ((r & m) == 0) {
        int q = r | m;
        bool ctl = bit8<CW>(lane, r) != 0;
        float t0r = sr[r], t0i = si[r], t1r = sr[q], t1i = si[q];
        sr[r] = ctl ? t1r : t0r;  si[r] = ctl ? t1i : t0i;
        sr[q] = ctl ? t0r : t1r;  si[q] = ctl ? t0i : t1i;
      }
    }
  }
}

// fused general single-qubit gate on distributed state, wire W in 0..3
// (cross-lane); U = 8 floats (u00,u01,u10,u11 as re,im pairs)
template <int W>
__device__ __forceinline__ void u8gate(float* sr, float* si, const float* __restrict__ U, int lane) {
  bool bit = ((lane >> (4 - W)) & 1) != 0;
  float u0 = U[0], u1 = U[1], u2 = U[2], u3 = U[3];
  float u4 = U[4], u5 = U[5], u6 = U[6], u7 = U[7];
  float car = bit ? u6 : u0, cai = bit ? u7 : u1;   // coeff of self
  float cpr = bit ? u4 : u2, cpi = bit ? u5 : u3;   // coeff of partner
  constexpr int mask = 16 >> W;
  #pragma unroll
  for (int r = 0; r < 8; ++r) {
    float pr = __shfl_xor(sr[r], mask, 32);
    float pi = __shfl_xor(si[r], mask, 32);
    float ar = sr[r], ai = si[r];
    sr[r] = car * ar - cai * ai + cpr * pr - cpi * pi;
    si[r] = car * ai + cai * ar + cpr * pi + cpi * pr;
  }
}

// ============ distributed 4-qubit value circuit (1 amp per lane, =====
// lanes 0-15, mirrored in lanes 16-31; wire w <-> bit 3-w of lane&15) ==
template <int W>
__device__ __forceinline__ void rxv(float& ar, float& ai, float c, float s) {
  constexpr int mask = 8 >> W;
  float pr = __shfl_xor(ar, mask, 32);
  float pi = __shfl_xor(ai, mask, 32);
  float a0r = ar, a0i = ai;
  ar = c * a0r + s * pi;
  ai = c * a0i - s * pr;
}

template <int W>
__device__ __forceinline__ void uvgate(float& ar, float& ai, const float* __restrict__ U, int l4) {
  bool bit = ((l4 >> (3 - W)) & 1) != 0;
  float car = bit ? U[6] : U[0], cai = bit ? U[7] : U[1];
  float cpr = bit ? U[4] : U[2], cpi = bit ? U[5] : U[3];
  constexpr int mask = 8 >> W;
  float pr = __shfl_xor(ar, mask, 32);
  float pi = __shfl_xor(ai, mask, 32);
  float a0r = ar, a0i = ai;
  ar = car * a0r - cai * a0i + cpr * pr - cpi * pi;
  ai = car * a0i + cai * a0r + cpr * pi + cpi * pr;
}

template <int C, int T>
__device__ __forceinline__ void crxv(float& ar, float& ai, float c, float s, int l4) {
  constexpr int mask = 8 >> T;
  float pr = __shfl_xor(ar, mask, 32);
  float pi = __shfl_xor(ai, mask, 32);
  bool ctl = ((l4 >> (3 - C)) & 1) != 0;
  float nr = c * ar + s * pi, ni = c * ai - s * pr;
  ar = ctl ? nr : ar;
  ai = ctl ? ni : ai;
}

template <int C, int T>
__device__ __forceinline__ void cnotv(float& ar, float& ai, int l4) {
  constexpr int mask = 8 >> T;
  float pr = __shfl_xor(ar, mask, 32);
  float pi = __shfl_xor(ai, mask, 32);
  bool ctl = ((l4 >> (3 - C)) & 1) != 0;
  ar = ctl ? pr : ar;
  ai = ctl ? pi : ai;
}

// =====================================================================
// Kernel 2: one sample per wave.
// =====================================================================
__global__ void qattn_sim(const float* __restrict__ xq, const float* __restrict__ xk,
                          const float* __restrict__ prep, const float* __restrict__ sg,
                          float* __restrict__ out, int nsamp) {
  int lane   = threadIdx.x & 31;
  int wave   = threadIdx.x >> 5;
  int sample = blockIdx.x * (blockDim.x >> 5) + wave;
  if (sample >= nsamp) return;   // whole-wave uniform exit

  float xqv[4], xkv[4];
  #pragma unroll
  for (int i = 0; i < 4; ++i) {
    xqv[i] = xq[(long)sample * 4 + i];
    xkv[i] = xk[(long)sample * 4 + i];
  }

  // ---- phase 1: parity-split factor states ----
  // even lanes build psi_q, odd lanes build psi_k, both as 16 amps in-lane.
  int par = lane & 1;
  float ar16[16], ai16[16];
  {
    const float* ubase = prep + (par ? P_UK : P_UQ);
    // per-wire 2-vectors v = U * Rx(x)|0>,  Rx(x)|0> = (cx, -i*sx)
    float w0r[4], w0i[4], w1r[4], w1i[4];
    #pragma unroll
    for (int i = 0; i < 4; ++i) {
      float x = par ? xkv[i] : xqv[i];
      float cx = __cosf(0.5f * x), sx = __sinf(0.5f * x);
      const float* U = ubase + 8 * i;
      w0r[i] = U[0] * cx + U[3] * sx;
      w0i[i] = U[1] * cx - U[2] * sx;
      w1r[i] = U[4] * cx + U[7] * sx;
      w1i[i] = U[5] * cx - U[6] * sx;
    }
    // product state: idx n, wire i <-> bit (3-i):  n = b0*8+b1*4+b2*2+b3
    float t4r[4], t4i[4], t8r[8], t8i[8];
    #pragma unroll
    for (int n = 0; n < 4; ++n) {
      float br = (n & 1) ? w1r[1] : w0r[1], bi = (n & 1) ? w1i[1] : w0i[1];
      float arv = (n >> 1) ? w1r[0] : w0r[0], aiv = (n >> 1) ? w1i[0] : w0i[0];
      t4r[n] = arv * br - aiv * bi;
      t4i[n] = arv * bi + aiv * br;
    }
    #pragma unroll
    for (int n = 0; n < 8; ++n) {
      float br = (n & 1) ? w1r[2] : w0r[2], bi = (n & 1) ? w1i[2] : w0i[2];
      t8r[n] = t4r[n >> 1] * br - t4i[n >> 1] * bi;
      t8i[n] = t4r[n >> 1] * bi + t4i[n >> 1] * br;
    }
    #pragma unroll
    for (int n = 0; n < 16; ++n) {
      float br = (n & 1) ? w1r[3] : w0r[3], bi = (n & 1) ? w1i[3] : w0i[3];
      ar16[n] = t8r[n >> 1] * br - t8i[n >> 1] * bi;
      ai16[n] = t8r[n >> 1] * bi + t8i[n >> 1] * br;
    }
    // entangling ring (in-lane)
    const float* cx4 = prep + (par ? P_CRXK : P_CRXQ);
    crx4<0,1>(ar16, ai16, cx4[0], cx4[1]);
    crx4<1,2>(ar16, ai16, cx4[2], cx4[3]);
    crx4<2,3>(ar16, ai16, cx4[4], cx4[5]);
    crx4<3,0>(ar16, ai16, cx4[6], cx4[7]);
    cnot4<0,1>(ar16, ai16); cnot4<1,2>(ar16, ai16);
    cnot4<2,3>(ar16, ai16); cnot4<3,0>(ar16, ai16);
  }

  // ---- reassemble distributed state: psi8[a] = psi_q[a>>4]*psi_k[a&15]
  // this lane's a = lane*8 + r -> q_idx = lane>>1, k_idx = par*8 + r
  float sr[8], si[8];
  {
    // kv[r]: odd lanes own psi_k[8+r]; even lanes fetch psi_k[r] from odd neighbor
    float kvr[8], kvi[8];
    #pragma unroll
    for (int r = 0; r < 8; ++r) {
      float tr = __shfl_xor(ar16[r], 1, 32);
      float ti = __shfl_xor(ai16[r], 1, 32);
      kvr[r] = par ? ar16[8 + r] : tr;
      kvi[r] = par ? ai16[8 + r] : ti;
    }
    // qv = psi_q[lane>>1]: select tree on own regs, then take even-neighbor's
    bool b0 = ((lane >> 1) & 1) != 0;
    bool b1 = ((lane >> 2) & 1) != 0;
    bool b2 = ((lane >> 3) & 1) != 0;
    bool b3 = ((lane >> 4) & 1) != 0;
    float u0r[8], u0i[8];
    #pragma unroll
    for (int j = 0; j < 8; ++j) {
      u0r[j] = b0 ? ar16[2*j+1] : ar16[2*j];
      u0i[j] = b0 ? ai16[2*j+1] : ai16[2*j];
    }
    float u1r[4], u1i[4];
    #pragma unroll
    for (int j = 0; j < 4; ++j) {
      u1r[j] = b1 ? u0r[2*j+1] : u0r[2*j];
      u1i[j] = b1 ? u0i[2*j+1] : u0i[2*j];
    }
    float u2r[2], u2i[2];
    #pragma unroll
    for (int j = 0; j < 2; ++j) {
      u2r[j] = b2 ? u1r[2*j+1] : u1r[2*j];
      u2i[j] = b2 ? u1i[2*j+1] : u1i[2*j];
    }
    float selr = b3 ? u2r[1] : u2r[0];
    float seli = b3 ? u2i[1] : u2i[0];
    float tr = __shfl_xor(selr, 1, 32);
    float ti = __shfl_xor(seli, 1, 32);
    float qvr = par ? tr : selr;
    float qvi = par ? ti : seli;
    #pragma unroll
    for (int r = 0; r < 8; ++r) {
      sr[r] = qvr * kvr[r] - qvi * kvi[r];
      si[r] = qvr * kvi[r] + qvi * kvr[r];
    }
  }

  // ---- phase 2: coupling + fused final rotations (distributed) ----
  const float* cc = prep + P_CRXC;
  crx8<0,4>(sr, si, cc[0],  cc[1],  lane);
  crx8<4,0>(sr, si, cc[8],  cc[9],  lane);
  crx8<1,5>(sr, si, cc[2],  cc[3],  lane);
  crx8<5,1>(sr, si, cc[10], cc[11], lane);
  crx8<2,6>(sr, si, cc[4],  cc[5],  lane);
  crx8<6,2>(sr, si, cc[12], cc[13], lane);
  crx8<3,7>(sr, si, cc[6],  cc[7],  lane);
  crx8<7,3>(sr, si, cc[14], cc[15], lane);
  cnot8<0,4>(sr, si, lane); cnot8<4,0>(sr, si, lane);
  cnot8<1,5>(sr, si, lane); cnot8<5,1>(sr, si, lane);
  cnot8<2,6>(sr, si, lane); cnot8<6,2>(sr, si, lane);
  cnot8<3,7>(sr, si, lane); cnot8<7,3>(sr, si, lane);
  u8gate<0>(sr, si, prep + P_UC + 0,  lane);
  u8gate<1>(sr, si, prep + P_UC + 8,  lane);
  u8gate<2>(sr, si, prep + P_UC + 16, lane);
  u8gate<3>(sr, si, prep + P_UC + 24, lane);

  // ---- expectations -> score ----
  float p = 0.f;
  #pragma unroll
  for (int r = 0; r < 8; ++r) p += sr[r] * sr[r] + si[r] * si[r];

  float score[4];
  #pragma unroll
  for (int w = 0; w < 4; ++w) {
    int lbit = (lane >> (4 - w)) & 1;
    float z = wave_sum(lbit ? -p : p);
    int mask = 16 >> w;
    float xloc = 0.f;
    #pragma unroll
    for (int r = 0; r < 8; ++r) {
      float pr = __shfl_xor(sr[r], mask, 32);
      float pi = __shfl_xor(si[r], mask, 32);
      xloc += sr[r] * pr + si[r] * pi;   // = 2*Re(sum conj(p0)*p1)
    }
    float xe = wave_sum(xloc);
    score[w] = sqrtf(z * z + xe * xe);
  }

  // ---- phase 3: value circuit, distributed 1 amp/lane (mirrored halves)
  int l4 = lane & 15;
  float var_ = (l4 == 0) ? 1.f : 0.f;
  float vai_ = 0.f;
  #pragma unroll
  for (int i = 0; i < 4; ++i) {
    float c = __cosf(0.5f * xkv[i]), s = __sinf(0.5f * xkv[i]);
    if (i == 0) rxv<0>(var_, vai_, c, s);
    if (i == 1) rxv<1>(var_, vai_, c, s);
    if (i == 2) rxv<2>(var_, vai_, c, s);
    if (i == 3) rxv<3>(var_, vai_, c, s);
  }
  uvgate<0>(var_, vai_, prep + P_UV + 0,  l4);
  uvgate<1>(var_, vai_, prep + P_UV + 8,  l4);
  uvgate<2>(var_, vai_, prep + P_UV + 16, l4);
  uvgate<3>(var_, vai_, prep + P_UV + 24, l4);
  const float* cv = prep + P_CRXV;
  crxv<0,1>(var_, vai_, cv[0], cv[1], l4);
  crxv<1,2>(var_, vai_, cv[2], cv[3], l4);
  crxv<2,3>(var_, vai_, cv[4], cv[5], l4);
  crxv<3,0>(var_, vai_, cv[6], cv[7], l4);
  cnotv<0,1>(var_, vai_, l4); cnotv<1,2>(var_, vai_, l4);
  cnotv<2,3>(var_, vai_, l4); cnotv<3,0>(var_, vai_, l4);
  #pragma unroll
  for (int i = 0; i < 4; ++i) {
    float ang = tanhf(score[i]) * sg[i];
    float c = __cosf(0.5f * ang), s = __sinf(0.5f * ang);
    if (i == 0) rxv<0>(var_, vai_, c, s);
    if (i == 1) rxv<1>(var_, vai_, c, s);
    if (i == 2) rxv<2>(var_, vai_, c, s);
    if (i == 3) rxv<3>(var_, vai_, c, s);
  }
  cnotv<0,1>(var_, vai_, l4); cnotv<1,2>(var_, vai_, l4);
  cnotv<2,3>(var_, vai_, l4); cnotv<3,0>(var_, vai_, l4);

  float pv = var_ * var_ + vai_ * vai_;
  float z2[4], x2[4];
  #pragma unroll
  for (int w = 0; w < 4; ++w) {
    int bit = (l4 >> (3 - w)) & 1;
    z2[w] = half_sum(bit ? -pv : pv);
    int mask = 8 >> w;
    float pr = __shfl_xor(var_, mask, 32);
    float pi = __shfl_xor(vai_, mask, 32);
    x2[w] = half_sum(var_ * pr + vai_ * pi);
  }

  if (lane == 0) {
    #pragma unroll
    for (int w = 0; w < 4; ++w) {
      out[(long)sample * 8 + w]     = z2[w];
      out[(long)sample * 8 + 4 + w] = x2[w];
    }
  }
}

extern "C" void kernel_launch(void* const* d_in, const int* in_sizes, int n_in,
                              void* d_out, int out_size, void* d_ws, size_t ws_size,
                              hipStream_t stream) {
  const float* x1     = (const float*)d_in[0];
  const float* x2     = (const float*)d_in[1];
  const float* W      = (const float*)d_in[2];
  const float* b      = (const float*)d_in[3];
  const float* wq_rot = (const float*)d_in[4];
  const float* wq_crx = (const float*)d_in[5];
  const float* wk_rot = (const float*)d_in[6];
  const float* wk_crx = (const float*)d_in[7];
  const float* wv_rot = (const float*)d_in[8];
  const float* wv_crx = (const float*)d_in[9];
  const float* wc_rot = (const float*)d_in[10];
  const float* wc_crx = (const float*)d_in[11];
  const float* sg     = (const float*)d_in[12];
  float* out = (float*)d_out;

  int B = in_sizes[0] / IN_DIM;              // 16384
  float* xq   = (float*)d_ws;                // [B,4]
  float* xk   = xq + (size_t)B * 4;          // [B,4]
  float* prep = xk + (size_t)B * 4;          // P_N floats

  qattn_prep<<<1, 64, 0, stream>>>(wq_rot, wq_crx, wk_rot, wk_crx,
                                   wv_rot, wv_crx, wc_rot, wc_crx, prep);

  int tiles = B / 16;
  qattn_proj<<<tiles / 8, 256, 0, stream>>>(x1, x2, W, b, xq, xk);

  qattn_sim<<<B / 8, 256, 0, stream>>>(xq, xk, prep, sg, out, B);
}